// GraphNN_79809082294964
// MI455X (gfx1250) — compile-verified
//
#include <hip/hip_runtime.h>

// ---------------------------------------------------------------------------
// Hetero-GraphSAGE (2 layers, mean aggregation, 4 edge types) for gfx1250.
//  - scatter-add aggregation with f32 global atomics (bandwidth-bound part),
//    software-pipelined with global_prefetch of the next edge's source row
//  - dense GEMMs on the WMMA fp32 path: v_wmma_f32_16x16x4_f32 (exact fp32)
//  - algebraic fold: sum_t x@Wr[t] = x@(sum_t Wr[t]) -> 5 GEMMs per layer
// ---------------------------------------------------------------------------

typedef __attribute__((ext_vector_type(2))) float v2f;
typedef __attribute__((ext_vector_type(8))) float v8f;

#define NTYPES 4

// ------------------------------ utility kernels ----------------------------

__global__ void zero_f32(float* __restrict__ p, long long n) {
  long long i = (long long)blockIdx.x * blockDim.x + threadIdx.x;
  long long stride = (long long)gridDim.x * blockDim.x;
  for (; i < n; i += stride) p[i] = 0.0f;
}

__global__ void count_edges(const int* __restrict__ dst, float* __restrict__ cnt, int E) {
  int i = blockIdx.x * blockDim.x + threadIdx.x;
  int stride = gridDim.x * blockDim.x;
  for (; i < E; i += stride) atomicAdd(&cnt[dst[i]], 1.0f);
}

__global__ void invert_cnt(float* __restrict__ c, int n) {
  int i = blockIdx.x * blockDim.x + threadIdx.x;
  if (i < n) c[i] = 1.0f / fmaxf(c[i], 1.0f);
}

// out[i] = sum over 4 types of in[i + t*n]   (weight/bias fold across types)
__global__ void sum_types(const float* __restrict__ in, float* __restrict__ out, long long n) {
  long long i = (long long)blockIdx.x * blockDim.x + threadIdx.x;
  long long stride = (long long)gridDim.x * blockDim.x;
  for (; i < n; i += stride)
    out[i] = in[i] + in[i + n] + in[i + 2 * n] + in[i + 3 * n];
}

// ------------------------- edge scatter (aggregation) ----------------------
// One wave per edge: coalesced float4 gather of the source row, float atomics
// into the destination row. D is 128 or 256 (multiple of 128).
// Software pipelining: lanes issue global_prefetch_b8 covering the *next*
// edge's source row while the current row is gathered/accumulated.
__global__ void scatter_add(const float* __restrict__ x, const int* __restrict__ src,
                            const int* __restrict__ dst, float* __restrict__ agg,
                            int E, int D) {
  int gwave = (int)((blockIdx.x * blockDim.x + threadIdx.x) >> 5);
  int lane  = threadIdx.x & 31;
  int nw    = (int)((gridDim.x * blockDim.x) >> 5);
  int chunks = D >> 7;            // float4-chunks per lane
  int pfstep = D >> 3;            // bytes per lane so 32 lanes span the row
  for (int e = gwave; e < E; e += nw) {
    int en = e + nw;
    if (en < E) {
      int sn = src[en];
      const char* pf = (const char*)(x + (size_t)sn * D) + lane * pfstep;
      __builtin_prefetch(pf, 0, 3);   // -> global_prefetch_b8 (speculative RT)
    }
    int s = src[e];
    int d = dst[e];
    const float4* xs = (const float4*)(x + (size_t)s * D);
    float* ad = agg + (size_t)d * D;
    for (int c = 0; c < chunks; ++c) {
      float4 v = xs[c * 32 + lane];
      int base = (c * 32 + lane) * 4;
      atomicAdd(ad + base + 0, v.x);
      atomicAdd(ad + base + 1, v.y);
      atomicAdd(ad + base + 2, v.z);
      atomicAdd(ad + base + 3, v.w);
    }
  }
}

// ------------------------------- WMMA GEMM ---------------------------------
// C[M x NC] (+)= scale(A)[M x K] @ B[K x NC]
//   block tile: 64(M) x 128(N), BK = 64; 256 threads = 8 waves (4 M x 2 N)
//   per wave: 1 x 4 subtiles of 16x16 -> 4 v8f accumulators
//   flags: bit0 = init (overwrite acc), bit1 = finalize (bias, *0.25, relu -> out)
__global__ __launch_bounds__(256) void gemm_wmma(
    const float* __restrict__ A, const float* __restrict__ scale,
    const float* __restrict__ B, float* __restrict__ acc,
    const float* __restrict__ bias, float* __restrict__ out,
    int M, int K, int NC, int flags) {
  __shared__ float As[64][66];    // padded: row stride 66 breaks bank aliasing
  __shared__ float Bs[64][132];   // padded: row stride 132

  const int tid  = threadIdx.x;
  const int wave = tid >> 5;
  const int lane = tid & 31;
  const int half = lane >> 4;     // 0: lanes 0-15, 1: lanes 16-31
  const int l16  = lane & 15;

  const int m0 = blockIdx.x * 64;
  const int n0 = blockIdx.y * 128;
  const int wm = wave >> 1;       // 0..3 -> wave's 16-row band
  const int wn = wave & 1;        // 0..1 -> wave's 64-col band
  const int mrow = wm * 16;
  const int ncol = wn * 64;

  v8f c[4];
  for (int nt = 0; nt < 4; ++nt)
    for (int v = 0; v < 8; ++v) c[nt][v] = 0.0f;

  // A-tile load mapping: 64 rows x 64 cols, 4 threads/row, 16 floats/thread
  const int ar = tid >> 2;
  const int ac = (tid & 3) * 16;
  // B-tile load mapping: 64 rows x 128 cols, 4 threads/row, 32 floats/thread
  const int br = tid >> 2;
  const int bc = (tid & 3) * 32;

  for (int k0 = 0; k0 < K; k0 += 64) {
    // ---- stage A (with optional 1/deg row scaling folded in) ----
    {
      const int r = m0 + ar;
      float s = 1.0f;
      if (scale != nullptr && r < M) s = scale[r];
      const float* Ap = A + (size_t)r * K + k0;
      for (int i = 0; i < 4; ++i) {
        int cc = ac + i * 4;
        float4 v = make_float4(0.f, 0.f, 0.f, 0.f);
        if (r < M) v = *(const float4*)(Ap + cc);
        As[ar][cc + 0] = v.x * s;
        As[ar][cc + 1] = v.y * s;
        As[ar][cc + 2] = v.z * s;
        As[ar][cc + 3] = v.w * s;
      }
    }
    // ---- stage B (K, NC are multiples of 64/128: no guards needed) ----
    {
      const float* Bp = B + (size_t)(k0 + br) * NC + n0 + bc;
      for (int i = 0; i < 8; ++i) {
        float4 v = *(const float4*)(Bp + i * 4);
        int cc = bc + i * 4;
        Bs[br][cc + 0] = v.x;
        Bs[br][cc + 1] = v.y;
        Bs[br][cc + 2] = v.z;
        Bs[br][cc + 3] = v.w;
      }
    }
    __syncthreads();

    // ---- 16 k-steps of K=4 fp32 WMMA ----
    for (int kk = 0; kk < 64; kk += 4) {
      const int ka = kk + 2 * half;  // half-wave K split per ISA A/B layout
      v2f a;
      a.x = As[mrow + l16][ka];
      a.y = As[mrow + l16][ka + 1];
#pragma unroll
      for (int nt = 0; nt < 4; ++nt) {
        v2f b;
        b.x = Bs[ka][ncol + nt * 16 + l16];
        b.y = Bs[ka + 1][ncol + nt * 16 + l16];
        c[nt] = __builtin_amdgcn_wmma_f32_16x16x4_f32(
            /*neg_a=*/false, a, /*neg_b=*/false, b,
            /*c_mod=*/(short)0, c[nt], /*reuse_a=*/false, /*reuse_b=*/false);
      }
    }
    __syncthreads();
  }

  // ---- epilogue: accumulate across GEMMs / fused bias+avg+relu ----
  const bool init = (flags & 1) != 0;
  const bool fin  = (flags & 2) != 0;
  for (int nt = 0; nt < 4; ++nt) {
    const int col = n0 + ncol + nt * 16 + l16;
    for (int v = 0; v < 8; ++v) {
      const int row = m0 + mrow + v + 8 * half;
      if (row < M) {
        const size_t idx = (size_t)row * NC + col;
        float r = c[nt][v];
        if (!init) r += acc[idx];
        if (fin) {
          r = (r + bias[col]) * 0.25f;   // HeteroConv 'mean' over 4 types
          out[idx] = fmaxf(r, 0.0f);     // ReLU
        } else {
          acc[idx] = r;
        }
      }
    }
  }
}

// ------------------------------- host driver -------------------------------

extern "C" void kernel_launch(void* const* d_in, const int* in_sizes, int n_in,
                              void* d_out, int out_size, void* d_ws, size_t ws_size,
                              hipStream_t stream) {
  const float* feats = (const float*)d_in[0];
  const int*   edges = (const int*)d_in[1];   // [T, 2, E]
  const float* Wl0   = (const float*)d_in[2];
  const float* Wr0   = (const float*)d_in[3];
  const float* b0    = (const float*)d_in[4];
  const float* Wl1   = (const float*)d_in[5];
  const float* Wr1   = (const float*)d_in[6];
  const float* b1    = (const float*)d_in[7];
  float* out = (float*)d_out;

  const int DF = 128;
  const int N  = in_sizes[0] / DF;                 // 100000
  const int E  = in_sizes[1] / (NTYPES * 2);       // 1000000
  const int H  = in_sizes[2] / (NTYPES * DF);      // 256
  const int EMB = in_sizes[5] / (NTYPES * H);      // 128

  // workspace layout (floats)
  float* ws   = (float*)d_ws;
  float* cnt  = ws;                                  // 4*N   inverse degrees
  float* agg  = cnt + (size_t)NTYPES * N;            // N*256 (reused per type)
  float* acc  = agg + (size_t)N * 256;               // N*256 GEMM accumulator
  float* h    = acc + (size_t)N * 256;               // N*256 hidden activations
  float* wsum = h   + (size_t)N * 256;               // 256*256 folded Wr
  float* bsum = wsum + 256 * 256;                    // 256 folded bias

  // per-type in-degree (same for both layers), then invert in place
  zero_f32<<<512, 256, 0, stream>>>(cnt, (long long)NTYPES * N);
  for (int t = 0; t < NTYPES; ++t)
    count_edges<<<1024, 256, 0, stream>>>(edges + (size_t)t * 2 * E + E,
                                          cnt + (size_t)t * N, E);
  invert_cnt<<<(NTYPES * N + 255) / 256, 256, 0, stream>>>(cnt, NTYPES * N);

  auto run_layer = [&](const float* xin, const float* Wl, const float* Wr,
                       const float* b, int K, int NC, float* obuf) {
    sum_types<<<(K * NC + 255) / 256, 256, 0, stream>>>(Wr, wsum, (long long)K * NC);
    sum_types<<<1, 256, 0, stream>>>(b, bsum, (long long)NC);
    dim3 ggrid((N + 63) / 64, NC / 128);
    for (int t = 0; t < NTYPES; ++t) {
      zero_f32<<<4096, 256, 0, stream>>>(agg, (long long)N * K);
      scatter_add<<<2048, 256, 0, stream>>>(xin,
                                            edges + (size_t)t * 2 * E,
                                            edges + (size_t)t * 2 * E + E,
                                            agg, E, K);
      // acc (+)= (agg * 1/deg) @ Wl[t]
      gemm_wmma<<<ggrid, 256, 0, stream>>>(agg, cnt + (size_t)t * N,
                                           Wl + (size_t)t * K * NC, acc,
                                           nullptr, nullptr, N, K, NC,
                                           t == 0 ? 1 : 0);
    }
    // obuf = relu((acc + x @ sum_t Wr[t] + sum_t b[t]) * 0.25)
    gemm_wmma<<<ggrid, 256, 0, stream>>>(xin, nullptr, wsum, acc,
                                         bsum, obuf, N, K, NC, 2);
  };

  run_layer(feats, Wl0, Wr0, b0, DF, H, h);   // layer 1: 128 -> 256
  run_layer(h,     Wl1, Wr1, b1, H, EMB, out); // layer 2: 256 -> 128
}